// Markowitz_6794638262272
// MI455X (gfx1250) — compile-verified
//
#include <hip/hip_runtime.h>

#define NDIM   512
#define BATCH  256
#define PITERS 30
#define FITERS 300
#define BISECT 24

typedef __attribute__((ext_vector_type(2))) float     v2f;
typedef __attribute__((ext_vector_type(8))) float     v8f;
typedef __attribute__((ext_vector_type(8))) _Float16  v8h;

// ---------------------------------------------------------------------------
// Kernel A: Q[b] = gamma[b]^2 * C[b]^T C[b]   via V_WMMA_F32_16X16X4_F32.
// Q stored as fp16 (128 MB total -> L2-resident for the GEMV phase).
// Block = 256 threads = 8 waves; wave grid 4(row)x2(col); each wave computes a
// 32x32 tile of Q as 2x2 WMMA tiles, K advanced 4 per step.
// A(m,k) = C[k][m], B(k,n) = C[k][n]  (Q = C^T C, symmetric).
// ---------------------------------------------------------------------------
__global__ __launch_bounds__(256) void markowitz_form_q(
    const float* __restrict__ cov,    // [B,N,N]
    const float* __restrict__ gamma,  // [B]
    _Float16*    __restrict__ Qh)     // [B,N,N] fp16
{
  const int b    = blockIdx.z;
  const int n0   = blockIdx.x * 64;    // column tile base
  const int m0   = blockIdx.y * 128;   // row tile base
  const int wave = threadIdx.x >> 5;
  const int lane = threadIdx.x & 31;
  const int rowBase = m0 + (wave & 3) * 32;
  const int colBase = n0 + (wave >> 2) * 32;

  const float g2 = gamma[b] * gamma[b];
  const float* C = cov + (size_t)b * NDIM * NDIM;

  // ISA A-operand layout (16x4 f32): lanes 0-15 -> (K0,K1), lanes 16-31 -> (K2,K3),
  // column = base + lane%16.  B-operand (4x16) mirrors it.
  const int la = lane & 15;
  const int lr = (lane >> 4) * 2;              // 0 or 2
  const float* pa0 = C + (size_t)lr * NDIM + rowBase      + la;
  const float* pa1 = C + (size_t)lr * NDIM + rowBase + 16 + la;
  const float* pb0 = C + (size_t)lr * NDIM + colBase      + la;
  const float* pb1 = C + (size_t)lr * NDIM + colBase + 16 + la;

  v8f acc00 = {}; v8f acc01 = {}; v8f acc10 = {}; v8f acc11 = {};

  #pragma unroll 2
  for (int k0 = 0; k0 < NDIM; k0 += 4) {
    const int o = k0 * NDIM;
    v2f a0; a0.x = pa0[o]; a0.y = pa0[o + NDIM];
    v2f a1; a1.x = pa1[o]; a1.y = pa1[o + NDIM];
    v2f b0; b0.x = pb0[o]; b0.y = pb0[o + NDIM];
    v2f b1; b1.x = pb1[o]; b1.y = pb1[o + NDIM];
    acc00 = __builtin_amdgcn_wmma_f32_16x16x4_f32(false, a0, false, b0, (short)0, acc00, false, false);
    acc01 = __builtin_amdgcn_wmma_f32_16x16x4_f32(false, a0, false, b1, (short)0, acc01, false, false);
    acc10 = __builtin_amdgcn_wmma_f32_16x16x4_f32(false, a1, false, b0, (short)0, acc10, false, false);
    acc11 = __builtin_amdgcn_wmma_f32_16x16x4_f32(false, a1, false, b1, (short)0, acc11, false, false);
  }

  // D layout (16x16 f32): VGPR r holds M = r + 8*(lane>=16), N = lane&15.
  _Float16* Qb = Qh + (size_t)b * NDIM * NDIM;
  const int dr = (lane >> 4) * 8;
  const int dc = lane & 15;
  #pragma unroll
  for (int r = 0; r < 8; ++r) {
    const int row0 = rowBase + r + dr;
    const int row1 = rowBase + 16 + r + dr;
    Qb[row0 * NDIM + colBase      + dc] = (_Float16)(g2 * acc00[r]);
    Qb[row0 * NDIM + colBase + 16 + dc] = (_Float16)(g2 * acc01[r]);
    Qb[row1 * NDIM + colBase      + dc] = (_Float16)(g2 * acc10[r]);
    Qb[row1 * NDIM + colBase + 16 + dc] = (_Float16)(g2 * acc11[r]);
  }
}

// ---------------------------------------------------------------------------
// Kernel B: fused power iteration + FISTA, one block (512 thr = 16 wave32) per
// batch. Q symmetric -> thread i reads ROW i contiguously with b128 fp16 loads.
// Simplex projection via fixed 24-step bisection (reductions, no sort).
// Block reductions: wave32 shfl_xor + double-buffered 16-slot LDS partials
// (one barrier per reduction; buffers alternate so last-read -> next-write of
// a buffer always has an intervening barrier).
// ---------------------------------------------------------------------------
__device__ inline float wave_sum32(float x) {
  #pragma unroll
  for (int off = 16; off > 0; off >>= 1) x += __shfl_xor(x, off, 32);
  return x;
}
__device__ inline float wave_max32(float x) {
  #pragma unroll
  for (int off = 16; off > 0; off >>= 1) x = fmaxf(x, __shfl_xor(x, off, 32));
  return x;
}
// One barrier; caller must alternate buffers between consecutive reductions.
__device__ inline float block_sum_1b(float x, float* buf) {
  x = wave_sum32(x);
  if ((threadIdx.x & 31) == 0) buf[threadIdx.x >> 5] = x;
  __syncthreads();
  const float4* b4 = (const float4*)buf;
  const float4 p0 = b4[0], p1 = b4[1], p2 = b4[2], p3 = b4[3];
  return ((p0.x + p0.y) + (p0.z + p0.w)) + ((p1.x + p1.y) + (p1.z + p1.w)) +
         ((p2.x + p2.y) + (p2.z + p2.w)) + ((p3.x + p3.y) + (p3.z + p3.w));
}
__device__ inline float block_max_1b(float x, float* buf) {
  x = wave_max32(x);
  if ((threadIdx.x & 31) == 0) buf[threadIdx.x >> 5] = x;
  __syncthreads();
  const float4* b4 = (const float4*)buf;
  const float4 p0 = b4[0], p1 = b4[1], p2 = b4[2], p3 = b4[3];
  float m = fmaxf(fmaxf(fmaxf(p0.x, p0.y), fmaxf(p0.z, p0.w)),
                  fmaxf(fmaxf(p1.x, p1.y), fmaxf(p1.z, p1.w)));
  m = fmaxf(m, fmaxf(fmaxf(fmaxf(p2.x, p2.y), fmaxf(p2.z, p2.w)),
                     fmaxf(fmaxf(p3.x, p3.y), fmaxf(p3.z, p3.w))));
  return m;
}

// Row-contiguous GEMV: 64 x global b128 (8 fp16) + 128 x LDS b128 per thread.
__device__ inline float qmatvec(const _Float16* __restrict__ qrow,
                                const float* __restrict__ yv) {
  float acc = 0.f;
  #pragma unroll 2
  for (int j = 0; j < NDIM; j += 8) {
    const v8h    q8 = *(const v8h*)(qrow + j);
    const float4 ya = *(const float4*)(yv + j);
    const float4 yb = *(const float4*)(yv + j + 4);
    acc += (float)q8[0] * ya.x;
    acc += (float)q8[1] * ya.y;
    acc += (float)q8[2] * ya.z;
    acc += (float)q8[3] * ya.w;
    acc += (float)q8[4] * yb.x;
    acc += (float)q8[5] * yb.y;
    acc += (float)q8[6] * yb.z;
    acc += (float)q8[7] * yb.w;
  }
  return acc;
}

__global__ __launch_bounds__(512) void markowitz_fista(
    const _Float16* __restrict__ Qh,   // [B,N,N] fp16
    const float*    __restrict__ rets, // [B,N]
    float*          __restrict__ out)  // [B,N]
{
  __shared__ __align__(16) float sh_y[NDIM];
  __shared__ __align__(16) float sbufA[16];
  __shared__ __align__(16) float sbufB[16];

  const int b = blockIdx.x;
  const int i = threadIdx.x;
  const _Float16* qrow = Qh + (size_t)b * NDIM * NDIM + i * NDIM;  // row i (symmetric Q)
  const float r = rets[b * NDIM + i];

  // ---- power iteration for lambda_max(Q) ----
  float v = 0.044194173824159216f;  // 1/sqrt(512)
  sh_y[i] = v;
  __syncthreads();
  float u = 0.f;
  for (int it = 0; it < PITERS; ++it) {
    u = qmatvec(qrow, sh_y);                 // reads sh_y
    const float s = block_sum_1b(u * u, sbufA);  // barrier: all matvec reads done
    v = u * (1.0f / (sqrtf(s) + 1e-12f));
    sh_y[i] = v;
    __syncthreads();                         // publish sh_y, retire sbufA reads
  }
  u = qmatvec(qrow, sh_y);
  const float lam = block_sum_1b(v * u, sbufA);
  const float eta = 1.0f / (2.0f * lam + 1e-8f);
  __syncthreads();                           // retire sh_y reads before rewrite

  // ---- FISTA on the simplex ----
  float w = 1.0f / NDIM, y = w, t = 1.0f;
  sh_y[i] = y;
  __syncthreads();
  for (int it = 0; it < FITERS; ++it) {
    const float gr = 2.0f * qmatvec(qrow, sh_y) - r;
    const float vv = y - eta * gr;

    // projection onto {w>=0, sum w = 1}: theta* in [max(v)-1, max(v)]
    const float vmax = block_max_1b(vv, sbufA);
    float lo = vmax - 1.0f, hi = vmax;
    #pragma unroll 1
    for (int bi = 0; bi < BISECT; ++bi) {
      const float th = 0.5f * (lo + hi);
      float* buf = (bi & 1) ? sbufA : sbufB;           // alternate buffers
      const float s = block_sum_1b(fmaxf(vv - th, 0.0f), buf);  // block-uniform
      if (s >= 1.0f) lo = th; else hi = th;
    }
    const float th = 0.5f * (lo + hi);
    const float wn = fmaxf(vv - th, 0.0f);

    const float tn = 0.5f * (1.0f + sqrtf(1.0f + 4.0f * t * t));
    const float yn = wn + ((t - 1.0f) / tn) * (wn - w);
    w = wn; t = tn; y = yn;
    sh_y[i] = yn;                 // all sh_y readers passed many barriers above
    __syncthreads();              // publish sh_y, retire last buffer reads
  }
  out[b * NDIM + i] = w;
}

// ---------------------------------------------------------------------------
extern "C" void kernel_launch(void* const* d_in, const int* in_sizes, int n_in,
                              void* d_out, int out_size, void* d_ws, size_t ws_size,
                              hipStream_t stream) {
  (void)in_sizes; (void)n_in; (void)out_size; (void)ws_size;
  const float* rets = (const float*)d_in[0];
  const float* cov  = (const float*)d_in[1];
  const float* gam  = (const float*)d_in[2];
  float* out        = (float*)d_out;
  _Float16* Qh      = (_Float16*)d_ws;   // needs B*N*N*2 = 128 MiB scratch

  dim3 gridA(NDIM / 64, NDIM / 128, BATCH);   // 8 x 4 x 256 blocks
  markowitz_form_q<<<gridA, dim3(256), 0, stream>>>(cov, gam, Qh);
  markowitz_fista<<<dim3(BATCH), dim3(NDIM), 0, stream>>>(Qh, rets, out);
}